// QValueAggregator_44384192037394
// MI455X (gfx1250) — compile-verified
//
#include <hip/hip_runtime.h>

typedef float v2f __attribute__((ext_vector_type(2)));
typedef float v8f __attribute__((ext_vector_type(8)));
typedef unsigned int u32x4 __attribute__((ext_vector_type(4)));
typedef int i32x4 __attribute__((ext_vector_type(4)));
typedef int i32x8 __attribute__((ext_vector_type(8)));

#define Bz 32
#define Lz 4096
#define Dz 512
#define ROWS_PER_WG 128          // 8 waves x 16 rows each
#define KC 32                    // K-chunk per TDM tile
#define LDS_STRIDE 36            // 32 + 4 pad (TDM pad_interval=32dw, pad_amount=4dw)
#define TILE_FLOATS (ROWS_PER_WG * LDS_STRIDE)

// Raw LDS byte offset of a __shared__ object (flat LDS addr low 32 bits = offset)
__device__ __forceinline__ unsigned lds_raw_offset(const void* p) {
  return (unsigned)(unsigned long long)
      (const __attribute__((address_space(3))) char*)p;
}

// ---------------------------------------------------------------------------
// TDM: async-load a [ROWS_PER_WG x KC] f32 tile (row stride Dz in memory)
// into LDS at ldsOff, with 4-DWORD padding after every 32 DWORDs so the
// LDS image has row stride LDS_STRIDE (bank-conflict-free WMMA A-feed).
// Descriptor per CDNA5 ISA D# groups 0/1 (2D tensor -> groups 2/3 zero).
// ---------------------------------------------------------------------------
__device__ __forceinline__ void tdm_load_tile(const float* gsrc, unsigned ldsOff) {
  const unsigned long long ga = (unsigned long long)(const void*)gsrc;
  u32x4 g0;
  g0[0] = 1u;                                               // count=1, user D#
  g0[1] = ldsOff;                                           // lds_addr (bytes)
  g0[2] = (unsigned)ga;                                     // global_addr[31:0]
  g0[3] = (unsigned)((ga >> 32) & 0x01FFFFFFu) | (2u << 30);// addr[56:32]|type=2
  i32x8 g1;
  g1[0] = (int)((2u << 16)      // data_size = 4 bytes
              | (1u << 20)      // pad_enable
              | (4u << 22)      // pad_interval: 32 DWORDs
              | (3u << 25));    // pad_amount: 4 DWORDs
  g1[1] = (int)(KC << 16);            // tensor_dim0 = KC     (bits 79:48 lo)
  g1[2] = (int)(ROWS_PER_WG << 16);   // tensor_dim1 = 128    (bits 111:80 lo)
  g1[3] = (int)(KC << 16);            // tile_dim0 = KC       (bits 127:112)
  g1[4] = ROWS_PER_WG;                // tile_dim1 = 128, tile_dim2 = 0
  g1[5] = Dz;                         // tensor_dim0_stride = 512
  g1[6] = 0;
  g1[7] = 0;
  const i32x4 z4 = {0, 0, 0, 0};
#if __clang_major__ >= 23
  const i32x8 z8 = {0, 0, 0, 0, 0, 0, 0, 0};
  __builtin_amdgcn_tensor_load_to_lds(g0, g1, z4, z4, z8, 0);
#else
  __builtin_amdgcn_tensor_load_to_lds(g0, g1, z4, z4, 0);
#endif
}

// ---------------------------------------------------------------------------
// Kernel 1: values[row] = dot(states[row,:], W) + b
//   TDM double-buffered LDS staging + V_WMMA_F32_16X16X4_F32 K-chain.
//   Wave w owns rows [w*16, w*16+16) of the workgroup's 128-row tile.
// ---------------------------------------------------------------------------
__global__ __launch_bounds__(256)
void qvalue_dot_wmma(const float* __restrict__ states,
                     const float* __restrict__ W,
                     const float* __restrict__ bias_p,
                     float* __restrict__ values) {
  __shared__ float sW[Dz];
  __shared__ float sTile[2 * TILE_FLOATS];

  const int      rowTile = blockIdx.x * ROWS_PER_WG;
  const unsigned tid  = threadIdx.x;
  const unsigned lane = tid & 31u;   // wave32
  const unsigned wv   = tid >> 5;    // 0..7

  // stage W (512 f32 = 2KB) once; published by first loop barrier
  for (int i = (int)tid; i < Dz; i += 256) sW[i] = W[i];

  // WMMA 16x4 f32 A-feed mapping:
  //   lanes 0-15 : K = k+0,k+1 of row m;  lanes 16-31 : K = k+2,k+3 of row m
  const int mrow  = (int)(lane & 15);
  const int khalf = (int)((lane >> 4) << 1);

  const float*   gbase    = states + (size_t)rowTile * Dz;
  const unsigned ldsTile0 = lds_raw_offset(&sTile[0]);
  const unsigned ldsTile1 = lds_raw_offset(&sTile[TILE_FLOATS]);

  v8f acc = {};

  if (wv == 0) tdm_load_tile(gbase, ldsTile0);           // prime buffer 0

  const int NCH = Dz / KC;                               // 16 chunks
  for (int c = 0; c < NCH; ++c) {
    if (wv == 0) {
      if (c + 1 < NCH) {
        // prefetch next chunk into the other buffer, then wait for current
        tdm_load_tile(gbase + (c + 1) * KC, (c & 1) ? ldsTile0 : ldsTile1);
        __builtin_amdgcn_s_wait_tensorcnt((short)1);     // oldest TDM done
      } else {
        __builtin_amdgcn_s_wait_tensorcnt((short)0);
      }
    }
    __syncthreads();                                     // tile c visible to all

    const float* aRow = &sTile[(c & 1) * TILE_FLOATS
                               + ((int)wv * 16 + mrow) * LDS_STRIDE];
    const float* wRow = &sW[c * KC];
    #pragma unroll
    for (int k = 0; k < KC; k += 4) {
      v2f a, bb;
      a.x  = aRow[k + khalf];
      a.y  = aRow[k + khalf + 1];
      bb.x = wRow[k + khalf];      // B[k][n] = W[k] for all 16 columns
      bb.y = wRow[k + khalf + 1];
      acc = __builtin_amdgcn_wmma_f32_16x16x4_f32(
          /*neg_a=*/false, a, /*neg_b=*/false, bb,
          /*c_mod=*/(short)0, acc, /*reuse_a=*/false, /*reuse_b=*/false);
    }
    __syncthreads();        // all readers done before this buffer is re-filled
  }

  // D layout: VGPR j -> M=j (lanes 0-15), M=j+8 (lanes 16-31); columns equal.
  const float bias = bias_p[0];
  if (lane == 0 || lane == 16) {
    const int base = rowTile + (int)wv * 16 + (lane ? 8 : 0);
    #pragma unroll
    for (int j = 0; j < 8; ++j) values[base + j] = acc[j] + bias;
  }
}

// ---------------------------------------------------------------------------
// Kernel 2: out[b,j] = sum_{i>=j} values[b,i]  (reverse cumsum, one WG per row)
// ---------------------------------------------------------------------------
__global__ __launch_bounds__(256)
void qvalue_suffix_scan(const float* __restrict__ values,
                        float* __restrict__ out) {
  __shared__ float s[256];
  const unsigned t = threadIdx.x;
  const size_t base = (size_t)blockIdx.x * Lz + (size_t)t * 16;

  float e[16];
  #pragma unroll
  for (int q = 0; q < 4; ++q) {
    const float4 v = *(const float4*)(values + base + q * 4);
    e[q*4+0] = v.x; e[q*4+1] = v.y; e[q*4+2] = v.z; e[q*4+3] = v.w;
  }

  float csum = 0.f;
  #pragma unroll
  for (int i = 0; i < 16; ++i) csum += e[i];
  s[t] = csum;
  __syncthreads();

  // Hillis-Steele inclusive suffix scan over 256 chunk sums
  #pragma unroll
  for (int off = 1; off < 256; off <<= 1) {
    const float v = (t + off < 256) ? s[t + off] : 0.f;
    __syncthreads();
    s[t] += v;
    __syncthreads();
  }

  const float offset = (t < 255) ? s[t + 1] : 0.f;
  float run = offset;
  float o[16];
  #pragma unroll
  for (int i = 15; i >= 0; --i) { run += e[i]; o[i] = run; }

  #pragma unroll
  for (int q = 0; q < 4; ++q) {
    float4 v; v.x = o[q*4]; v.y = o[q*4+1]; v.z = o[q*4+2]; v.w = o[q*4+3];
    *(float4*)(out + base + q * 4) = v;
  }
}

// ---------------------------------------------------------------------------
extern "C" void kernel_launch(void* const* d_in, const int* in_sizes, int n_in,
                              void* d_out, int out_size, void* d_ws, size_t ws_size,
                              hipStream_t stream) {
  (void)in_sizes; (void)n_in; (void)out_size; (void)ws_size;
  const float* states = (const float*)d_in[0];
  const float* W      = (const float*)d_in[1];
  const float* b      = (const float*)d_in[2];
  float* values = (float*)d_ws;          // B*L f32 = 512 KB scratch
  float* out    = (float*)d_out;

  const int nRowTiles = (Bz * Lz) / ROWS_PER_WG;   // 1024
  qvalue_dot_wmma<<<dim3(nRowTiles), dim3(256), 0, stream>>>(states, W, b, values);
  qvalue_suffix_scan<<<dim3(Bz), dim3(256), 0, stream>>>(values, out);
}